// TGCNGraphConvolution_36601711297136
// MI455X (gfx1250) — compile-verified
//
#include <hip/hip_runtime.h>
#include <hip/hip_bf16.h>

typedef __attribute__((ext_vector_type(2))) float v2f;
typedef __attribute__((ext_vector_type(8))) float v8f;

static constexpr int NN = 16384;   // nodes
static constexpr int BB = 2;       // batch
static constexpr int FF = 32;      // features
static constexpr int GG = 16;      // gru units
static constexpr int OO = 16;      // output dim
static constexpr int CC = FF + GG; // 48
static constexpr int NWAVES = 8;
static constexpr int KCH = NN / NWAVES;  // 2048 K per wave
static constexpr int MT  = 2;            // M-tiles (16 rows each) per CTA

// ---------------------------------------------------------------------------
// Kernel 1: cwT[b][o][n] = sum_c cat[b][n][c] * W[c][o]   (transposed store;
// bias is applied after the Laplacian multiply: (L@cat)@W + b = L@(cat@W) + b).
// idx = b*2^18 + o*2^14 + n  ->  flat idx IS the [B,O,N] offset, and
// consecutive threads vary n, so the store is fully coalesced.
// ---------------------------------------------------------------------------
__global__ __launch_bounds__(256) void tgcn_catw(
    const float* __restrict__ x,   // [B,N,F]
    const float* __restrict__ h,   // [B,N*G]
    const float* __restrict__ W,   // [C,O]
    float* __restrict__ cwT)       // [B,O,N]
{
    __shared__ float sW[CC * OO];
    const int t = threadIdx.x;
    for (int i = t; i < CC * OO; i += 256) sW[i] = W[i];
    __syncthreads();

    const int idx = blockIdx.x * 256 + t;
    const int n = idx & (NN - 1);
    const int o = (idx >> 14) & (OO - 1);
    const int b = idx >> 18;

    const float* __restrict__ xr = x + (size_t)(b * NN + n) * FF;
    const float* __restrict__ hr = h + (size_t)(b * NN + n) * GG;

    float s = 0.f;
#pragma unroll
    for (int f = 0; f < FF; ++f) s += xr[f] * sW[f * OO + o];
#pragma unroll
    for (int g = 0; g < GG; ++g) s += hr[g] * sW[(FF + g) * OO + o];
    cwT[idx] = s;
}

// ---------------------------------------------------------------------------
// Kernel 2: out[b] = L @ cw[b] + bias, 32 rows (2 M-tiles) per CTA.
// 8 waves split K=16384; per K-step each wave does 4 b64 loads + 4
// v_wmma_f32_16x16x4_f32 (2 M-tiles x 2 batches).
//   A frag (16x4 f32):  vgpr0 = L[m, k+2hi], vgpr1 = L[m, k+2hi+1]
//   B frag (4x16 f32):  vgpr0 = cwT[o=hl][k+2hi], vgpr1 = cwT[hl][k+2hi+1]
// L is streamed once -> non-temporal loads; cwT (2 MB) stays L2-resident.
// Cross-wave reduction via LDS (row stride 20 -> conflict-free), bias, store.
// ---------------------------------------------------------------------------
__global__ __launch_bounds__(256) void tgcn_lap_gemm(
    const float* __restrict__ L,    // [N,N]
    const float* __restrict__ cwT,  // [B,O,N]
    const float* __restrict__ bias, // [O]
    float* __restrict__ out)        // [B,N*O]
{
    __shared__ float red[NWAVES][MT][BB][16][20];  // 40 KB

    const int t    = threadIdx.x;
    const int w    = t >> 5;
    const int lane = t & 31;
    const int hl   = lane & 15;   // M (A) / O-column (B) within tile
    const int hi   = lane >> 4;   // half-wave: K offset 0 vs 2
    const int m_base = blockIdx.x * (MT * 16);

    const int k0 = w * KCH;

    // A pointers: one per M-tile, offset to this lane's row + K phase
    const float* __restrict__ A0 = L + (size_t)(m_base + hl) * NN + 2 * hi;
    const float* __restrict__ A1 = A0 + (size_t)16 * NN;
    // B pointers: one per batch; lane hl owns row o=hl of cwT
    const float* __restrict__ B0 = cwT + (size_t)hl * NN + 2 * hi;
    const float* __restrict__ B1 = B0 + (size_t)OO * NN;

    v8f acc00 = {0.f, 0.f, 0.f, 0.f, 0.f, 0.f, 0.f, 0.f};  // mtile0, batch0
    v8f acc01 = {0.f, 0.f, 0.f, 0.f, 0.f, 0.f, 0.f, 0.f};  // mtile0, batch1
    v8f acc10 = {0.f, 0.f, 0.f, 0.f, 0.f, 0.f, 0.f, 0.f};  // mtile1, batch0
    v8f acc11 = {0.f, 0.f, 0.f, 0.f, 0.f, 0.f, 0.f, 0.f};  // mtile1, batch1

#pragma unroll 4
    for (int k = k0; k < k0 + KCH; k += 4) {
        v2f a0 = __builtin_nontemporal_load((const v2f*)(A0 + k));
        v2f a1 = __builtin_nontemporal_load((const v2f*)(A1 + k));
        v2f b0 = *(const v2f*)(B0 + k);
        v2f b1 = *(const v2f*)(B1 + k);

        acc00 = __builtin_amdgcn_wmma_f32_16x16x4_f32(
            false, a0, false, b0, (short)0, acc00, false, false);
        acc01 = __builtin_amdgcn_wmma_f32_16x16x4_f32(
            false, a0, false, b1, (short)0, acc01, false, false);
        acc10 = __builtin_amdgcn_wmma_f32_16x16x4_f32(
            false, a1, false, b0, (short)0, acc10, false, false);
        acc11 = __builtin_amdgcn_wmma_f32_16x16x4_f32(
            false, a1, false, b1, (short)0, acc11, false, false);
    }

    // C/D layout: VGPR v -> (M = v + 8*hi, N = hl). Spill for K reduction.
#pragma unroll
    for (int v = 0; v < 8; ++v) {
        const int m = v + 8 * hi;
        red[w][0][0][m][hl] = acc00[v];
        red[w][0][1][m][hl] = acc01[v];
        red[w][1][0][m][hl] = acc10[v];
        red[w][1][1][m][hl] = acc11[v];
    }
    __syncthreads();

    // 256 threads -> one (m,n) element each; loop over M-tiles and batches.
    const int m = t >> 4;
    const int n = t & 15;
    const float bn = bias[n];
#pragma unroll
    for (int mt = 0; mt < MT; ++mt) {
#pragma unroll
        for (int b = 0; b < BB; ++b) {
            float s = 0.f;
#pragma unroll
            for (int ww = 0; ww < NWAVES; ++ww) s += red[ww][mt][b][m][n];
            s += bn;
            out[(size_t)b * NN * OO + (size_t)(m_base + mt * 16 + m) * OO + n] = s;
        }
    }
}

// ---------------------------------------------------------------------------
extern "C" void kernel_launch(void* const* d_in, const int* in_sizes, int n_in,
                              void* d_out, int out_size, void* d_ws, size_t ws_size,
                              hipStream_t stream) {
    const float* x    = (const float*)d_in[0];  // inputs       [B,N,F]
    const float* h    = (const float*)d_in[1];  // hidden_state [B,N*G]
    const float* L    = (const float*)d_in[2];  // laplacian    [N,N]
    const float* W    = (const float*)d_in[3];  // weights      [C,O]
    const float* bias = (const float*)d_in[4];  // biases       [O]
    float* out = (float*)d_out;                 // [B, N*O] f32
    float* cwT = (float*)d_ws;                  // scratch: [B,O,N] = 2 MB

    tgcn_catw<<<(BB * NN * OO) / 256, 256, 0, stream>>>(x, h, W, cwT);
    tgcn_lap_gemm<<<NN / (MT * 16), 256, 0, stream>>>(L, cwT, bias, out);
}